// VectorQuantizer_10763188044143
// MI455X (gfx1250) — compile-verified
//
#include <hip/hip_runtime.h>
#include <hip/hip_bf16.h>

typedef __attribute__((ext_vector_type(2))) float    v2f;
typedef __attribute__((ext_vector_type(8))) float    v8f;
typedef __attribute__((ext_vector_type(4))) int      v4i;
typedef __attribute__((ext_vector_type(8))) int      v8i;
typedef __attribute__((ext_vector_type(4))) unsigned v4u;

#define N_EMB   1024
#define DIM     256
#define NROWS   65536        // 16 * 64 * 64
#define HWSZ    4096         // 64*64
#define WAVES_PER_BLOCK 8    // 256 threads; 8 waves share one LDS-staged tile
#define NWAVES  (NROWS / 16) // 4096 waves, 16 rows each
#define TCODES  32           // codes per staged tile (two 16-wide WMMA tiles)
#define LROW    260          // 256 + 4 pad DWORDs -> conflict-free banks, TDM pad match

// --- staging path selection: TDM > async-to-LDS > plain --------------------
#define USE_TDM        0
#define USE_ASYNC_LDS  0
#if defined(__has_builtin)
#  if __has_builtin(__builtin_amdgcn_tensor_load_to_lds) && \
      __has_builtin(__builtin_amdgcn_s_wait_tensorcnt)
#    undef  USE_TDM
#    define USE_TDM 1
#  elif __has_builtin(__builtin_amdgcn_global_load_async_to_lds_b128)
#    undef  USE_ASYNC_LDS
#    define USE_ASYNC_LDS 1
#  endif
#endif

typedef v4i __attribute__((address_space(1)))* as1_v4i_p;  // global
typedef v4i __attribute__((address_space(3)))* as3_v4i_p;  // LDS

__device__ __forceinline__ as1_v4i_p to_glb(const void* p) {
  return (as1_v4i_p)(unsigned long long)(uintptr_t)p;
}
__device__ __forceinline__ as3_v4i_p to_lds(void* p) {
  // LDS flat address: low 32 bits are the LDS byte offset.
  return (as3_v4i_p)(unsigned)(uintptr_t)p;
}

// Stage one TCODES x 256 f32 codebook tile into LDS (row-major, LROW stride).
__device__ __forceinline__ void stage_tile(const float* __restrict__ emb,
                                           int c0, float* dst, int t) {
#if USE_TDM
  // One TDM descriptor per tile, issued by wave 0 only (uniform branch).
  if (t < 32) {
    const unsigned long long ga = (unsigned long long)(uintptr_t)(emb + (size_t)c0 * DIM);
    const unsigned lds = (unsigned)(uintptr_t)dst;
    // D# group 0: count=1 | lds_addr | global_addr(57b) | type=2
    v4u g0;
    g0[0] = 1u;                                        // count=1, user mode
    g0[1] = lds;                                       // LDS byte address
    g0[2] = (unsigned)ga;                              // global_addr[31:0]
    g0[3] = (unsigned)((ga >> 32) & 0x01FFFFFFu) | 0x80000000u; // [56:32] | type=2
    // D# group 1: data_size=4B, pad_enable, pad every 256 DW by 4 DW,
    // tensor 256 x 1024 f32 (stride 256), tile 256 x 32.
    v8i g1;
    g1[0] = (int)((2u << 16) | (1u << 20) | (7u << 22) | (3u << 25));
    g1[1] = (int)(256u << 16);                 // tensor_dim0[15:0]=256 @bits[63:48]
    g1[2] = (int)((1024u & 0xFFFFu) << 16);    // dim0 hi=0 | tensor_dim1 lo=1024
    g1[3] = (int)(256u << 16);                 // dim1 hi=0 | tile_dim0=256
    g1[4] = (int)(32u);                        // tile_dim1=32 | tile_dim2=0
    g1[5] = (int)(256u);                       // tensor_dim0_stride lo32 = 256
    g1[6] = (int)(256u << 16);                 // stride0 hi=0 | tensor_dim1_stride lo=256
    g1[7] = 0;                                 // tensor_dim1_stride hi
    v4i zero4 = {};
    v8i zero8 = {};
    __builtin_amdgcn_tensor_load_to_lds(g0, g1, zero4, zero4, zero8, 0);
  }
#else
  // 256 threads: 8 threads per code row, 8 float4 (32 floats) per thread.
  const int code = t >> 3;          // 0..31
  const int base = (t & 7) * 4;     // float offset within row
  const float* src = emb + (size_t)(c0 + code) * DIM + base;
  float* d = dst + code * LROW + base;
#pragma unroll
  for (int i = 0; i < 8; ++i) {
#  if USE_ASYNC_LDS
    __builtin_amdgcn_global_load_async_to_lds_b128(
        to_glb(src + i * 32), to_lds(d + i * 32), 0, 0);
#  else
    *(float4*)(d + i * 32) = *(const float4*)(src + i * 32);
#  endif
  }
#endif
}

// Make staged data visible: per-path counter wait; caller then __syncthreads().
__device__ __forceinline__ void stage_fence(int t) {
#if USE_TDM
  if (t < 32) __builtin_amdgcn_s_wait_tensorcnt(0);
#elif USE_ASYNC_LDS
#  if defined(__has_builtin) && __has_builtin(__builtin_amdgcn_s_wait_asynccnt)
  __builtin_amdgcn_s_wait_asynccnt(0);
#  else
  asm volatile("s_wait_asynccnt 0x0" ::: "memory");
#  endif
#else
  (void)t;
#endif
}

// ---------------------------------------------------------------------------
// Kernel 1: per-code squared norms ||e_k||^2 -> cnorm[1024]
// ---------------------------------------------------------------------------
__global__ void __launch_bounds__(256) vq_norms(const float* __restrict__ emb,
                                                float* __restrict__ cnorm) {
  int k = blockIdx.x * blockDim.x + threadIdx.x;
  if (k < N_EMB) {
    const float4* r = (const float4*)(emb + (size_t)k * DIM);
    float s = 0.f;
#pragma unroll
    for (int i = 0; i < DIM / 4; ++i) {
      float4 v = r[i];
      s += v.x * v.x + v.y * v.y + v.z * v.z + v.w * v.w;
    }
    cnorm[k] = s;
  }
}

// ---------------------------------------------------------------------------
// Kernel 2: WMMA distances + argmin + gather + loss partials.
// 8 waves / block; each wave owns 16 rows (A strip resident in 128 VGPRs).
// Codebook swept in 32-code tiles double-buffered through LDS (TDM/async);
// two independent WMMA accumulator chains per wave for matrix-pipe ILP.
// ---------------------------------------------------------------------------
__global__ void __launch_bounds__(WAVES_PER_BLOCK * 32)
vq_main(const float* __restrict__ z, const float* __restrict__ emb,
        const float* __restrict__ cnorm, float* __restrict__ out,
        float* __restrict__ partial) {
  __shared__ float btile[2][TCODES * LROW];
  __shared__ int   sidx[WAVES_PER_BLOCK * 16];

  const int t      = threadIdx.x;
  const int lane   = t & 31;
  const int wave   = t >> 5;
  const int waveId = blockIdx.x * WAVES_PER_BLOCK + wave; // 0..4095
  const int row0   = waveId * 16;
  const int b      = row0 >> 12;
  const int hw0    = row0 & (HWSZ - 1);
  const int m      = lane & 15;       // row (A) / code-in-tile (B) for this lane
  const int kb     = (lane >> 4) * 2; // K sub-offset within each 4-wide step

  // z element (row0+m, dim d) lives at z[(b*256 + d)*4096 + hw0 + m]
  const float* zb = z + ((size_t)b * DIM) * HWSZ + (size_t)(hw0 + m);

  // Preload the full 16x256 A strip: fragment f covers dims 4f..4f+3.
  v2f a[DIM / 4];
#pragma unroll
  for (int f = 0; f < DIM / 4; ++f) {
    int d = 4 * f + kb;
    a[f].x = zb[(size_t)d * HWSZ];
    a[f].y = zb[(size_t)(d + 1) * HWSZ];
  }

  float best[8];
  int   bidx[8];
#pragma unroll
  for (int r = 0; r < 8; ++r) { best[r] = 3.4e38f; bidx[r] = 0; }

  // Prologue: stage first tile.
  stage_tile(emb, 0, btile[0], t);
  stage_fence(t);
  __syncthreads();

  int buf = 0;
  for (int c0 = 0; c0 < N_EMB; c0 += TCODES, buf ^= 1) {
    // Prefetch next tile into the other buffer (overlaps with WMMAs below).
    if (c0 + TCODES < N_EMB) stage_tile(emb, c0 + TCODES, btile[buf ^ 1], t);

    v8f acc0 = {};
    v8f acc1 = {};
    const float* b0 = &btile[buf][m * LROW + kb];           // codes c0..c0+15
    const float* b1 = b0 + 16 * LROW;                       // codes c0+16..c0+31
#pragma unroll
    for (int f = 0; f < DIM / 4; ++f) {
      v2f bf0 = *(const v2f*)(b0 + 4 * f);
      v2f bf1 = *(const v2f*)(b1 + 4 * f);
      acc0 = __builtin_amdgcn_wmma_f32_16x16x4_f32(false, a[f], false, bf0,
                                                   (short)0, acc0, false, false);
      acc1 = __builtin_amdgcn_wmma_f32_16x16x4_f32(false, a[f], false, bf1,
                                                   (short)0, acc1, false, false);
    }

    float cn0 = cnorm[c0 + m];
    float cn1 = cnorm[c0 + 16 + m];
#pragma unroll
    for (int r = 0; r < 8; ++r) {
      // dist_k minus the per-row constant ||z||^2:  ||e||^2 - 2 z.e
      float s0 = cn0 - 2.0f * acc0[r];
      float s1 = cn1 - 2.0f * acc1[r];
      if (s0 < best[r]) { best[r] = s0; bidx[r] = c0 + m; }       // lower index first
      if (s1 < best[r]) { best[r] = s1; bidx[r] = c0 + 16 + m; }
    }

    stage_fence(t);     // next tile landed in LDS
    __syncthreads();    // everyone done reading btile[buf] before it is reused
  }

  // Min-reduce (value, index) across the 16 lanes of each half.
  // Lanes 0-15 hold rows 0..7 in acc slots, lanes 16-31 hold rows 8..15.
#pragma unroll
  for (int mask = 1; mask <= 8; mask <<= 1) {
#pragma unroll
    for (int r = 0; r < 8; ++r) {
      float ov = __shfl_xor(best[r], mask, 32);
      int   oi = __shfl_xor(bidx[r], mask, 32);
      if (ov < best[r] || (ov == best[r] && oi < bidx[r])) {
        best[r] = ov; bidx[r] = oi;
      }
    }
  }

  // Broadcast winning indices (16 per wave) through LDS.
  if (m == 0) {  // lanes 0 and 16 hold the reduced results
    int base = wave * 16 + (lane >> 4) * 8;
#pragma unroll
    for (int r = 0; r < 8; ++r) sidx[base + r] = bidx[r];
  }
  __syncthreads();
  const int myidx = sidx[wave * 16 + m];

  // Output phase: z_q_out[b][d][hw0+m] = emb[myidx][d]; reuse a[] for z.
  const float* erow = emb + (size_t)myidx * DIM;
  float* obase = out + 1 + ((size_t)b * DIM) * HWSZ + (size_t)(hw0 + m);
  float ssum = 0.f;
#pragma unroll
  for (int f = 0; f < DIM / 4; ++f) {
    int d = 4 * f + kb;
    v2f e = *(const v2f*)(erow + d);          // 8B-aligned (kb even), one b64
    obase[(size_t)d * HWSZ]       = e.x;
    obase[(size_t)(d + 1) * HWSZ] = e.y;
    float d0 = e.x - a[f].x;
    float d1 = e.y - a[f].y;
    ssum += d0 * d0 + d1 * d1;
  }

  // Deterministic per-wave loss partial.
#pragma unroll
  for (int mask = 1; mask <= 16; mask <<= 1)
    ssum += __shfl_xor(ssum, mask, 32);
  if (lane == 0) partial[waveId] = ssum;
}

// ---------------------------------------------------------------------------
// Kernel 3: fixed-order reduction of wave partials -> d_out[0]
// ---------------------------------------------------------------------------
__global__ void __launch_bounds__(256) vq_finish(const float* __restrict__ partial,
                                                 float* __restrict__ out) {
  __shared__ float s[256];
  float v = 0.f;
  for (int i = threadIdx.x; i < NWAVES; i += 256) v += partial[i];
  s[threadIdx.x] = v;
  __syncthreads();
  for (int w = 128; w > 0; w >>= 1) {
    if ((int)threadIdx.x < w) s[threadIdx.x] += s[threadIdx.x + w];
    __syncthreads();
  }
  if (threadIdx.x == 0)
    out[0] = s[0] * (1.0f / (16.0f * 256.0f * 64.0f * 64.0f));
}

// ---------------------------------------------------------------------------
extern "C" void kernel_launch(void* const* d_in, const int* in_sizes, int n_in,
                              void* d_out, int out_size, void* d_ws, size_t ws_size,
                              hipStream_t stream) {
  const float* z   = (const float*)d_in[0];   // [16,256,64,64] f32
  const float* emb = (const float*)d_in[1];   // [1024,256] f32
  float* out = (float*)d_out;                 // [1 + 16*256*64*64] f32
  float* ws  = (float*)d_ws;
  float* cnorm   = ws;            // 1024 floats
  float* partial = ws + N_EMB;    // 4096 floats

  vq_norms<<<N_EMB / 256, 256, 0, stream>>>(emb, cnorm);
  vq_main<<<NROWS / (WAVES_PER_BLOCK * 16), WAVES_PER_BLOCK * 32, 0, stream>>>(
      z, emb, cnorm, out, partial);
  vq_finish<<<1, 256, 0, stream>>>(partial, out);
}